// GlimpseSensor_45767171506547
// MI455X (gfx1250) — compile-verified
//
#include <hip/hip_runtime.h>

// GlimpseSensor for MI455X (gfx1250, wave32).
//
// rho[b,p,i,j,c] = images[b, fy, fx, c],  fy = fix0[b] + 2^p*(i-8),
//                                         fx = fix1[b] + 2^p*(j-8),
//                  zero outside [0,512)   (zero padding in the reference)
// fix[b] = rint((loc_normd[b]+1)/2 * 511)   (RNE == jnp.round)
//
// Pure gather (~1.2 MB traffic, ~50 ns of HBM time) -> latency / launch bound.
// Single kernel, 2D grid (p = blockIdx.x in {0,1,2}, b = blockIdx.y in 0..63)
// so no integer division; 256 threads = 8 wave32s per block, lane t = (i,j).
// Packed 12-byte loads (global_load_b96); non-temporal merged stores for the
// write-once output. p==0 blocks also emit the 2 fixation values (lane 0).

#define IMG_H 512
#define G_W   16
#define K_SC  3   // scales
#define RHO_ELEMS (64 * K_SC * G_W * G_W * 3)

struct __attribute__((packed, aligned(4))) f3 { float r, g, b; };

__global__ __launch_bounds__(256) void glimpse_kernel(
    const float* __restrict__ images,   // (64, 512, 512, 3) f32
    const float* __restrict__ loc,      // (64, 2) f32 in [-1,1]
    float* __restrict__ out)            // rho (64,3,16,16,3) f32, then fix (64,2)
{
    const int p  = blockIdx.x;          // scale 0..2
    const int b  = blockIdx.y;          // image 0..63
    const int t  = threadIdx.x;         // 0..255
    const int i  = t >> 4;              // patch row
    const int j  = t & 15;              // patch col

    // Fixation: uniform per block -> scalar load + broadcast math.
    const float l0 = loc[2 * b + 0];
    const float l1 = loc[2 * b + 1];
    const int fy0 = (int)rintf((l0 + 1.0f) * 0.5f * (float)(IMG_H - 1));
    const int fx0 = (int)rintf((l1 + 1.0f) * 0.5f * (float)(IMG_H - 1));

    const int y = fy0 + ((i - 8) << p);
    const int x = fx0 + ((j - 8) << p);

    f3 v; v.r = 0.0f; v.g = 0.0f; v.b = 0.0f;
    if (((unsigned)y < (unsigned)IMG_H) & ((unsigned)x < (unsigned)IMG_H)) {
        // 12-byte packed load -> global_load_b96 (4-byte aligned, DWORD mode)
        const f3* src = (const f3*)(images + (((size_t)b * IMG_H + y) * IMG_H + x) * 3);
        v = *src;
    }

    // Block writes a contiguous 3072-byte span; lane t owns [t*12, t*12+12).
    // Non-temporal: write-once output, keep caches for shared image lines.
    float* dst = out + ((size_t)(b * K_SC + p) * (G_W * G_W) + t) * 3;
    __builtin_nontemporal_store(v.r, dst + 0);
    __builtin_nontemporal_store(v.g, dst + 1);
    __builtin_nontemporal_store(v.b, dst + 2);

    // Fixations tail (as f32; values are exact small ints). One lane per image.
    if ((p == 0) & (t == 0)) {
        float* fx_out = out + RHO_ELEMS + 2 * b;
        __builtin_nontemporal_store((float)fy0, fx_out + 0);
        __builtin_nontemporal_store((float)fx0, fx_out + 1);
    }
}

extern "C" void kernel_launch(void* const* d_in, const int* in_sizes, int n_in,
                              void* d_out, int out_size, void* d_ws, size_t ws_size,
                              hipStream_t stream) {
    const float* images = (const float*)d_in[0];   // (64,512,512,3) f32
    const float* loc    = (const float*)d_in[1];   // (64,2) f32
    float* out = (float*)d_out;                    // rho flat, then fixations

    // grid (3 scales, 64 images): p = blockIdx.x, b = blockIdx.y -> no div/mod
    glimpse_kernel<<<dim3(K_SC, 64), 256, 0, stream>>>(images, loc, out);
}